// ICC_9861244911783
// MI455X (gfx1250) — compile-verified
//
#include <hip/hip_runtime.h>
#include <hip/hip_bf16.h>
#include <math.h>

typedef float v2f __attribute__((ext_vector_type(2)));
typedef float v8f __attribute__((ext_vector_type(8)));

#define RESV      0.5f
#define INV_RES   2.0f            // 1/RES
#define N_VOX     65536
#define M_PTS     2048
#define JTILES    (M_PTS / 16)    // 128 column tiles
#define BLK2      256             // 8 waves per block
#define ROWS_PER_BLK 128          // 8 waves * 16 rows
#define NBLK2     (N_VOX / ROWS_PER_BLK) // 512

// ---------------------------------------------------------------------------
// 4x4 inverse (adjugate). Layout-agnostic (row-major in -> row-major inverse).
// ---------------------------------------------------------------------------
__device__ inline void invert4(const float* m, float* o) {
    float inv[16];
    inv[0]  =  m[5]*m[10]*m[15] - m[5]*m[11]*m[14] - m[9]*m[6]*m[15] + m[9]*m[7]*m[14] + m[13]*m[6]*m[11] - m[13]*m[7]*m[10];
    inv[4]  = -m[4]*m[10]*m[15] + m[4]*m[11]*m[14] + m[8]*m[6]*m[15] - m[8]*m[7]*m[14] - m[12]*m[6]*m[11] + m[12]*m[7]*m[10];
    inv[8]  =  m[4]*m[9]*m[15]  - m[4]*m[11]*m[13] - m[8]*m[5]*m[15] + m[8]*m[7]*m[13] + m[12]*m[5]*m[11] - m[12]*m[7]*m[9];
    inv[12] = -m[4]*m[9]*m[14]  + m[4]*m[10]*m[13] + m[8]*m[5]*m[14] - m[8]*m[6]*m[13] - m[12]*m[5]*m[10] + m[12]*m[6]*m[9];
    inv[1]  = -m[1]*m[10]*m[15] + m[1]*m[11]*m[14] + m[9]*m[2]*m[15] - m[9]*m[3]*m[14] - m[13]*m[2]*m[11] + m[13]*m[3]*m[10];
    inv[5]  =  m[0]*m[10]*m[15] - m[0]*m[11]*m[14] - m[8]*m[2]*m[15] + m[8]*m[3]*m[14] + m[12]*m[2]*m[11] - m[12]*m[3]*m[10];
    inv[9]  = -m[0]*m[9]*m[15]  + m[0]*m[11]*m[13] + m[8]*m[1]*m[15] - m[8]*m[3]*m[13] - m[12]*m[1]*m[11] + m[12]*m[3]*m[9];
    inv[13] =  m[0]*m[9]*m[14]  - m[0]*m[10]*m[13] - m[8]*m[1]*m[14] + m[8]*m[2]*m[13] + m[12]*m[1]*m[10] - m[12]*m[2]*m[9];
    inv[2]  =  m[1]*m[6]*m[15]  - m[1]*m[7]*m[14]  - m[5]*m[2]*m[15] + m[5]*m[3]*m[14] + m[13]*m[2]*m[7]  - m[13]*m[3]*m[6];
    inv[6]  = -m[0]*m[6]*m[15]  + m[0]*m[7]*m[14]  + m[4]*m[2]*m[15] - m[4]*m[3]*m[14] - m[12]*m[2]*m[7]  + m[12]*m[3]*m[6];
    inv[10] =  m[0]*m[5]*m[15]  - m[0]*m[7]*m[13]  - m[4]*m[1]*m[15] + m[4]*m[3]*m[13] + m[12]*m[1]*m[7]  - m[12]*m[3]*m[5];
    inv[14] = -m[0]*m[5]*m[14]  + m[0]*m[6]*m[13]  + m[4]*m[1]*m[14] - m[4]*m[2]*m[13] - m[12]*m[1]*m[6]  + m[12]*m[2]*m[5];
    inv[3]  = -m[1]*m[6]*m[11]  + m[1]*m[7]*m[10]  + m[5]*m[2]*m[11] - m[5]*m[3]*m[10] - m[9]*m[2]*m[7]   + m[9]*m[3]*m[6];
    inv[7]  =  m[0]*m[6]*m[11]  - m[0]*m[7]*m[10]  - m[4]*m[2]*m[11] + m[4]*m[3]*m[10] + m[8]*m[2]*m[7]   - m[8]*m[3]*m[6];
    inv[11] = -m[0]*m[5]*m[11]  + m[0]*m[7]*m[9]   + m[4]*m[1]*m[11] - m[4]*m[3]*m[9]  - m[8]*m[1]*m[7]   + m[8]*m[3]*m[5];
    inv[15] =  m[0]*m[5]*m[10]  - m[0]*m[6]*m[9]   - m[4]*m[1]*m[10] + m[4]*m[2]*m[9]  + m[8]*m[1]*m[6]   - m[8]*m[2]*m[5];
    float det = m[0]*inv[0] + m[1]*inv[4] + m[2]*inv[8] + m[3]*inv[12];
    det = 1.0f / det;
    for (int i = 0; i < 16; ++i) o[i] = inv[i] * det;
}

// ---------------------------------------------------------------------------
// Kernel 1: transform model points, emit B-panel: bvec[j] = (-2p, |p|^2)
// in voxel coordinates (pred/RES).
// ---------------------------------------------------------------------------
__global__ void k_prep(const float* __restrict__ quat, const float* __restrict__ tran,
                       const float* __restrict__ model, const float* __restrict__ view,
                       float4* __restrict__ bvec) {
    int j = blockIdx.x * blockDim.x + threadIdx.x;
    if (j >= M_PTS) return;

    // quaternion_matrix(quat, tran)
    float q0 = quat[0], q1 = quat[1], q2 = quat[2], q3 = quat[3];
    float s  = sqrtf(2.0f / (q0*q0 + q1*q1 + q2*q2 + q3*q3));
    float w = q0*s, x = q1*s, y = q2*s, z = q3*s;
    float Q11 = x*x, Q22 = y*y, Q33 = z*z;
    float Q12 = x*y, Q13 = x*z, Q23 = y*z;
    float Q10 = x*w, Q20 = y*w, Q30 = z*w;
    float R[16] = {
        1.0f-Q22-Q33, Q12-Q30,      Q13+Q20,      tran[0],
        Q12+Q30,      1.0f-Q11-Q33, Q23-Q10,      tran[1],
        Q13-Q20,      Q23+Q10,      1.0f-Q11-Q22, tran[2],
        0.0f,         0.0f,         0.0f,         1.0f };

    float V[16];
    #pragma unroll
    for (int i = 0; i < 16; ++i) V[i] = view[i];
    float Vi[16];
    invert4(V, Vi);

    float W[16];
    #pragma unroll
    for (int r = 0; r < 4; ++r)
        #pragma unroll
        for (int c = 0; c < 4; ++c)
            W[r*4+c] = Vi[r*4+0]*R[0*4+c] + Vi[r*4+1]*R[1*4+c]
                     + Vi[r*4+2]*R[2*4+c] + Vi[r*4+3]*R[3*4+c];

    float mx = model[j*3+0], my = model[j*3+1], mz = model[j*3+2];
    float px = W[0]*mx  + W[1]*my  + W[2]*mz  + W[3];
    float py = W[4]*mx  + W[5]*my  + W[6]*mz  + W[7];
    float pz = W[8]*mx  + W[9]*my  + W[10]*mz + W[11];
    float pw = W[12]*mx + W[13]*my + W[14]*mz + W[15];
    float iw = 1.0f / pw;
    px = px * iw * INV_RES;   // pred / RES
    py = py * iw * INV_RES;
    pz = pz * iw * INV_RES;
    bvec[j] = make_float4(-2.0f*px, -2.0f*py, -2.0f*pz, px*px + py*py + pz*pz);
}

// ---------------------------------------------------------------------------
// Kernel 2: per wave, one 16-row strip of d2 = |c'|^2 + (A x B) via WMMA
// f32 16x16x4, min-reduced over all 2048 columns. Block partial sums out.
// ---------------------------------------------------------------------------
__global__ void __launch_bounds__(BLK2)
k_main(const float* __restrict__ centers, const float* __restrict__ freev,
       const float* __restrict__ oov, const float* __restrict__ masks,
       const float4* __restrict__ bvec, float* __restrict__ partials) {
    __shared__ float4 sB[M_PTS];        // 32 KB B-panel
    __shared__ float  sPart[16*3];

    for (int i = threadIdx.x; i < M_PTS; i += BLK2) sB[i] = bvec[i];
    __syncthreads();

    const int lane = threadIdx.x & 31;
    const int wave = threadIdx.x >> 5;
    const int hi   = lane >> 4;          // 0: K=0..1 half, 1: K=2..3 half
    const int lrow = lane & 15;
    const int i0   = blockIdx.x * ROWS_PER_BLK + wave * 16;
    const int row  = i0 + lrow;

    // A fragment (16x4 f32): lanes 0-15 hold K=0,1; lanes 16-31 hold K=2,3
    float cx = centers[row*3+0] * INV_RES;
    float cy = centers[row*3+1] * INV_RES;
    float cz = centers[row*3+2] * INV_RES;
    float cn2 = cx*cx + cy*cy + cz*cz;
    v2f a;
    a.x = hi ? cz : cx;
    a.y = hi ? 1.0f : cy;

    const float* sBf = reinterpret_cast<const float*>(sB);

    v8f mf;
    #pragma unroll
    for (int k = 0; k < 8; ++k) mf[k] = 3.0e38f;

    for (int jt = 0; jt < JTILES; ++jt) {
        int j = jt * 16 + lrow;
        // B fragment (4x16 f32): lanes 0-15 rows K=0,1; lanes 16-31 rows K=2,3
        const float2 bb = *reinterpret_cast<const float2*>(&sBf[j*4 + hi*2]);
        v2f b; b.x = bb.x; b.y = bb.y;
        v8f c = {};
        // D = A*B + 0 : (-2 c.p + |p|^2) for a 16x16 tile
        v8f d = __builtin_amdgcn_wmma_f32_16x16x4_f32(
            /*neg_a=*/false, a, /*neg_b=*/false, b,
            /*c_mod=*/(short)0, c, /*reuse_a=*/false, /*reuse_b=*/false);
        #pragma unroll
        for (int k = 0; k < 8; ++k) mf[k] = fminf(mf[k], d[k]);
    }

    // min across the 16 lanes holding the same row (columns within tile)
    #pragma unroll
    for (int k = 0; k < 8; ++k) {
        float v = mf[k];
        v = fminf(v, __shfl_xor(v, 8, 32));
        v = fminf(v, __shfl_xor(v, 4, 32));
        v = fminf(v, __shfl_xor(v, 2, 32));
        v = fminf(v, __shfl_xor(v, 1, 32));
        mf[k] = v;
    }

    // gather |c'|^2 for the rows this lane's accumulator VGPRs represent:
    // VGPR r <-> row i0 + r + 8*hi, whose cn2 lives in lane (r + 8*hi)
    float cn2r[8];
    #pragma unroll
    for (int r = 0; r < 8; ++r) cn2r[r] = __shfl(cn2, r + hi*8, 32);

    float psum = 0.0f, wsum = 0.0f, msum = 0.0f;
    if (lrow == 0) {
        #pragma unroll
        for (int r = 0; r < 8; ++r) {
            float d2   = mf[r] + cn2r[r];
            float dmin = sqrtf(fmaxf(d2, 0.0f));
            dmin = fminf(dmin, RESV * 0.5f);
            float occ = fmaxf(1.0f - dmin * (2.0f / RESV), 0.0f);
            int rr = i0 + r + hi*8;
            psum += occ;
            wsum += (freev[rr] + oov[rr]) * occ;
            msum += masks[rr] * occ;
        }
        int slot = wave * 2 + hi;        // 16 slots per block
        sPart[slot*3+0] = psum;
        sPart[slot*3+1] = wsum;
        sPart[slot*3+2] = msum;
    }
    __syncthreads();
    if (threadIdx.x == 0) {              // fixed-order sum -> deterministic
        float p = 0.0f, w = 0.0f, m = 0.0f;
        #pragma unroll
        for (int s2 = 0; s2 < 16; ++s2) {
            p += sPart[s2*3+0]; w += sPart[s2*3+1]; m += sPart[s2*3+2];
        }
        partials[blockIdx.x*3+0] = p;
        partials[blockIdx.x*3+1] = w;
        partials[blockIdx.x*3+2] = m;
    }
}

// ---------------------------------------------------------------------------
// Kernel 3: deterministic final reduction over 512 block partials + mask_sum
// ---------------------------------------------------------------------------
__global__ void __launch_bounds__(512)
k_final(const float* __restrict__ partials, const float* __restrict__ masks,
        float* __restrict__ out) {
    __shared__ float sp[512], sw[512], sm[512], sk[512];
    int t = threadIdx.x;
    sp[t] = partials[t*3+0];
    sw[t] = partials[t*3+1];
    sm[t] = partials[t*3+2];
    float k = 0.0f;
    for (int i = t*128; i < t*128 + 128; ++i) k += masks[i];
    sk[t] = k;
    __syncthreads();
    for (int s = 256; s > 0; s >>= 1) {
        if (t < s) {
            sp[t] += sp[t+s]; sw[t] += sw[t+s];
            sm[t] += sm[t+s]; sk[t] += sk[t+s];
        }
        __syncthreads();
    }
    if (t == 0) {
        float pred_sum = sp[0], wsum = sw[0], msum = sm[0], mask_sum = sk[0];
        float t1 = (pred_sum > 0.0f) ? (wsum / pred_sum) : 0.0f;
        float t2 = (mask_sum > 0.0f) ? (msum / mask_sum) : 0.0f;
        out[0] = t1 - t2;
    }
}

// ---------------------------------------------------------------------------
extern "C" void kernel_launch(void* const* d_in, const int* in_sizes, int n_in,
                              void* d_out, int out_size, void* d_ws, size_t ws_size,
                              hipStream_t stream) {
    const float* quat    = (const float*)d_in[0];
    const float* tran    = (const float*)d_in[1];
    const float* model   = (const float*)d_in[2];
    const float* view    = (const float*)d_in[3];
    const float* centers = (const float*)d_in[4];
    const float* freev   = (const float*)d_in[5];
    const float* oov     = (const float*)d_in[6];
    const float* masks   = (const float*)d_in[7];
    float* out = (float*)d_out;

    float4* bvec     = (float4*)d_ws;                       // 2048 * 16 B = 32 KB
    float*  partials = (float*)d_ws + M_PTS * 4;            // 512 * 3 floats

    k_prep<<<(M_PTS + 255) / 256, 256, 0, stream>>>(quat, tran, model, view, bvec);
    k_main<<<NBLK2, BLK2, 0, stream>>>(centers, freev, oov, masks, bvec, partials);
    k_final<<<1, 512, 0, stream>>>(partials, masks, out);
}